// QAOA_GNN_Router_69148973466104
// MI455X (gfx1250) — compile-verified
//
#include <hip/hip_runtime.h>
#include <hip/hip_bf16.h>

typedef __attribute__((ext_vector_type(2))) float v2f;
typedef __attribute__((ext_vector_type(8))) float v8f;

// ---------------------------------------------------------------------------
// K0: zero a float buffer (degree accumulator must be re-zeroed every call)
// ---------------------------------------------------------------------------
__global__ void k_zero_f32(float* __restrict__ p, int n) {
    int i = blockIdx.x * blockDim.x + threadIdx.x;
    if (i < n) p[i] = 0.0f;
}

// ---------------------------------------------------------------------------
// K1: in-degree via L2 atomics (deg counts dst occurrences; self-loop is the
//     +1 inside rsqrt later).
// ---------------------------------------------------------------------------
__global__ void k_degree(const long long* __restrict__ dst,
                         float* __restrict__ deg, int E) {
    int e = blockIdx.x * blockDim.x + threadIdx.x;
    if (e >= E) return;
    __builtin_prefetch(&dst[e + 8192], 0, 1);   // global_prefetch_b8
    atomicAdd(&deg[(int)dst[e]], 1.0f);
}

// ---------------------------------------------------------------------------
// K2: dinv = rsqrt(deg + 1)  (in place; always > 0 due to self loop)
// ---------------------------------------------------------------------------
__global__ void k_dinv(float* __restrict__ degdinv, int n) {
    int i = blockIdx.x * blockDim.x + threadIdx.x;
    if (i < n) degdinv[i] = rsqrtf(degdinv[i] + 1.0f);
}

// ---------------------------------------------------------------------------
// K3: h1 = x @ W1 via V_WMMA_F32_16X16X4_F32 (exact fp32 path).
//     One wave = 16 nodes. A: 16x4 (feature 3 zero-padded), B: W1 padded to
//     4x16 (cols 8..15 zero via mask-multiply, NOT branches — EXEC must stay
//     all-ones through the WMMA and we want select-based codegen).
//     Writes h1 and out1 := h1 * dinv^2 (self-loop term = accumulator init).
//     A layout: lane L -> row M=L&15, K-pair {0,1} (L<16) / {2,3} (L>=16).
//     D layout: VGPR v -> row base + v + 8*(L>=16), col = L&15.
// ---------------------------------------------------------------------------
__global__ void k_h1_wmma(const float* __restrict__ x,
                          const float* __restrict__ W1,   // [3][8]
                          const float* __restrict__ dinv,
                          float* __restrict__ h1,         // [N][8]
                          float* __restrict__ out1,       // [N][8]
                          int N) {
    int lane = threadIdx.x & 31;
    int wave = (blockIdx.x * blockDim.x + threadIdx.x) >> 5;
    int base = wave * 16;
    if (base >= N) return;                 // wave-uniform exit

    int half = lane >> 4;                  // 0: K={0,1}  1: K={2,3}
    int m    = lane & 15;
    int row  = base + m; if (row >= N) row = N - 1;   // clamp (tail safety)

    // A operand — computed addresses + selects, no divergent branches.
    const float* xr = x + (size_t)row * 3;
    v2f a;
    a.x = xr[half * 2];                    // f0 or f2
    a.y = half ? 0.0f : xr[1];             // f1 or pad

    // B operand — column n of padded W1; zero via mask multiply.
    int   n     = m;
    int   nc    = n & 7;
    float nmask = (n < 8) ? 1.0f : 0.0f;
    v2f b;
    b.x = W1[half * 16 + nc] * nmask;      // W1[0][n] or W1[2][n]
    b.y = half ? 0.0f : W1[8 + nc] * nmask;// W1[1][n] or pad row k=3

    v8f c = {};
    c = __builtin_amdgcn_wmma_f32_16x16x4_f32(
            false, a, false, b, (short)0, c, false, false);

    if (n < 8) {
        size_t rb = (size_t)base + (size_t)half * 8;   // first row this lane owns
        float*       hp = h1   + rb * 8 + n;
        float*       op = out1 + rb * 8 + n;
        const float* dp = dinv + rb;
        if (base + 16 <= N) {
            // Fast path: straight-line, offset-immediate stores.
#pragma unroll
            for (int v = 0; v < 8; ++v) {
                float hv = c[v];
                float dv = dp[v];
                hp[(size_t)v * 8] = hv;
                op[(size_t)v * 8] = hv * dv * dv;
            }
        } else {
#pragma unroll
            for (int v = 0; v < 8; ++v) {
                if ((int)rb + v < N) {
                    float hv = c[v];
                    float dv = dp[v];
                    hp[(size_t)v * 8] = hv;
                    op[(size_t)v * 8] = hv * dv * dv;
                }
            }
        }
    }
}

// ---------------------------------------------------------------------------
// K4: layer-1 gather-scale-scatter. 1 thread / edge, 8 channels.
//     out1[d] += h1[s] * dinv[s]*dinv[d]   (L2-resident f32 atomics)
// ---------------------------------------------------------------------------
__global__ void k_edge_l1(const long long* __restrict__ src,
                          const long long* __restrict__ dst,
                          const float* __restrict__ dinv,
                          const float* __restrict__ h1,
                          float* __restrict__ out1, int E) {
    int e = blockIdx.x * blockDim.x + threadIdx.x;
    if (e >= E) return;
    __builtin_prefetch(&src[e + 8192], 0, 1);
    __builtin_prefetch(&dst[e + 8192], 0, 1);
    int s = (int)src[e];
    int d = (int)dst[e];
    float nrm = dinv[s] * dinv[d];
    const float4* hs = (const float4*)(h1 + (size_t)s * 8);
    float4 h0 = hs[0];
    float4 h4 = hs[1];
    float* o = out1 + (size_t)d * 8;
    atomicAdd(&o[0], h0.x * nrm);
    atomicAdd(&o[1], h0.y * nrm);
    atomicAdd(&o[2], h0.z * nrm);
    atomicAdd(&o[3], h0.w * nrm);
    atomicAdd(&o[4], h4.x * nrm);
    atomicAdd(&o[5], h4.y * nrm);
    atomicAdd(&o[6], h4.z * nrm);
    atomicAdd(&o[7], h4.w * nrm);
}

// ---------------------------------------------------------------------------
// K5: h2 = relu(out1 + b1) @ W2 via two chained f32 WMMAs (K=8 = 2 x K=4).
//     Writes h2 [N][2]; initializes d_out = b2 + h2*dinv^2 (bias + self-loop
//     folded) ready for layer-2 atomic accumulation.
// ---------------------------------------------------------------------------
__global__ void k_h2_wmma(const float* __restrict__ out1,  // [N][8]
                          const float* __restrict__ W2,    // [8][2]
                          const float* __restrict__ b1,    // [8]
                          const float* __restrict__ b2,    // [2]
                          const float* __restrict__ dinv,
                          float* __restrict__ h2,          // [N][2]
                          float* __restrict__ out2,        // d_out [N][2]
                          int N) {
    int lane = threadIdx.x & 31;
    int wave = (blockIdx.x * blockDim.x + threadIdx.x) >> 5;
    int base = wave * 16;
    if (base >= N) return;                 // wave-uniform exit

    int half = lane >> 4;
    int m    = lane & 15;
    int row  = base + m; if (row >= N) row = N - 1;

    int ka = half * 2;        // K-chunk 0: k = {0,1} | {2,3}
    int kb = ka + 4;          // K-chunk 1: k = {4,5} | {6,7}
    const float* orow = out1 + (size_t)row * 8;

    v2f a0, a1;
    a0.x = fmaxf(orow[ka + 0] + b1[ka + 0], 0.0f);
    a0.y = fmaxf(orow[ka + 1] + b1[ka + 1], 0.0f);
    a1.x = fmaxf(orow[kb + 0] + b1[kb + 0], 0.0f);
    a1.y = fmaxf(orow[kb + 1] + b1[kb + 1], 0.0f);

    int   n     = m;
    int   nc    = n & 1;
    float nmask = (n < 2) ? 1.0f : 0.0f;
    v2f w0, w1;
    w0.x = W2[(ka + 0) * 2 + nc] * nmask;
    w0.y = W2[(ka + 1) * 2 + nc] * nmask;
    w1.x = W2[(kb + 0) * 2 + nc] * nmask;
    w1.y = W2[(kb + 1) * 2 + nc] * nmask;

    v8f c = {};
    c = __builtin_amdgcn_wmma_f32_16x16x4_f32(false, a0, false, w0,
                                              (short)0, c, false, false);
    c = __builtin_amdgcn_wmma_f32_16x16x4_f32(false, a1, false, w1,
                                              (short)0, c, false, false);

    if (n < 2) {
        size_t rb = (size_t)base + (size_t)half * 8;
        float*       hp = h2   + rb * 2 + n;
        float*       op = out2 + rb * 2 + n;
        const float* dp = dinv + rb;
        float bias = b2[n];
        if (base + 16 <= N) {
#pragma unroll
            for (int v = 0; v < 8; ++v) {
                float hv = c[v];
                float dv = dp[v];
                hp[(size_t)v * 2] = hv;
                op[(size_t)v * 2] = bias + hv * dv * dv;
            }
        } else {
#pragma unroll
            for (int v = 0; v < 8; ++v) {
                if ((int)rb + v < N) {
                    float hv = c[v];
                    float dv = dp[v];
                    hp[(size_t)v * 2] = hv;
                    op[(size_t)v * 2] = bias + hv * dv * dv;
                }
            }
        }
    }
}

// ---------------------------------------------------------------------------
// K6: layer-2 gather-scale-scatter, 2 channels, straight into d_out.
// ---------------------------------------------------------------------------
__global__ void k_edge_l2(const long long* __restrict__ src,
                          const long long* __restrict__ dst,
                          const float* __restrict__ dinv,
                          const float* __restrict__ h2,
                          float* __restrict__ out2, int E) {
    int e = blockIdx.x * blockDim.x + threadIdx.x;
    if (e >= E) return;
    __builtin_prefetch(&src[e + 8192], 0, 1);
    __builtin_prefetch(&dst[e + 8192], 0, 1);
    int s = (int)src[e];
    int d = (int)dst[e];
    float nrm = dinv[s] * dinv[d];
    float2 hv = *(const float2*)(h2 + (size_t)s * 2);
    atomicAdd(&out2[(size_t)d * 2 + 0], hv.x * nrm);
    atomicAdd(&out2[(size_t)d * 2 + 1], hv.y * nrm);
}

// ---------------------------------------------------------------------------
extern "C" void kernel_launch(void* const* d_in, const int* in_sizes, int n_in,
                              void* d_out, int out_size, void* d_ws, size_t ws_size,
                              hipStream_t stream) {
    const float*     x  = (const float*)d_in[0];
    const long long* ei = (const long long*)d_in[1];   // int64 [2][E]
    const float*     W1 = (const float*)d_in[2];
    const float*     b1 = (const float*)d_in[3];
    const float*     W2 = (const float*)d_in[4];
    const float*     b2 = (const float*)d_in[5];
    float*           out = (float*)d_out;

    const int N = in_sizes[0] / 3;
    const int E = in_sizes[1] / 2;
    const long long* src = ei;
    const long long* dst = ei + E;

    // workspace: dinv[N] | h1[N*8] | out1[N*8] | h2[N*2]   (~38 MB)
    float* ws   = (float*)d_ws;
    float* dinv = ws;
    float* h1   = dinv + N;
    float* out1 = h1 + (size_t)N * 8;
    float* h2   = out1 + (size_t)N * 8;

    const int B = 256;
    const int gN = (N + B - 1) / B;
    const int gE = (E + B - 1) / B;
    const int waves = (N + 15) / 16;          // 16 nodes per wave32
    const int gW = (waves + 7) / 8;           // 8 waves per 256-thread block

    k_zero_f32<<<gN, B, 0, stream>>>(dinv, N);
    k_degree  <<<gE, B, 0, stream>>>(dst, dinv, E);
    k_dinv    <<<gN, B, 0, stream>>>(dinv, N);
    k_h1_wmma <<<gW, B, 0, stream>>>(x, W1, dinv, h1, out1, N);
    k_edge_l1 <<<gE, B, 0, stream>>>(src, dst, dinv, h1, out1, E);
    k_h2_wmma <<<gW, B, 0, stream>>>(out1, W2, b1, b2, dinv, h2, out, N);
    k_edge_l2 <<<gE, B, 0, stream>>>(src, dst, dinv, h2, out, E);
}